// EPGSimulationMT_GPU_Vectorized_12781822672905
// MI455X (gfx1250) — compile-verified
//
#include <hip/hip_runtime.h>
#include <hip/hip_bf16.h>
#include <math.h>

// EPG-MT simulation, CDNA5 / gfx1250 (wave32).
//
// Round-2 design: each voxel is split across a LANE PAIR of a wave32:
//   even lane: EPG orders g = 0..10   (local slots 0..10)
//   odd  lane: EPG orders g = 11..20  (local slots 0..9; slot 10 = dummy,
//              forced to zero every pulse so nothing leaks via the Fm shift)
// -> 16384 lanes / 512 waves (2x occupancy, half the serial chain vs 1 lane
//    per voxel), 6 __shfl_xor lane-pair exchanges per pulse for the EPG shift
//    boundary, zero control divergence.
// State lives in VGPRs, ping-ponged between two register array sets so the
// EPG shift is folded into the write indices (no register-move pass).
// Fpb/Fmb are provably always zero in this reference -> stored as constants.
// Output (881 MB, > 192 MB L2) is streamed with non-temporal stores; the
// backend merges the contiguous v2f stores into global_store_b128 nt.

#define N_STATES 21
#define LS 11  // local slots per lane

typedef float v2f __attribute__((ext_vector_type(2)));

// ---- per-pulse RF scalars (computed by both lanes of a pair, uniform) ----
#define EPG_PULSE_CONSTS(pidx)                                              \
    const float fa   = flip[(pidx)];                                        \
    const float ph   = phase[(pidx)];                                       \
    const float hal  = 0.5f * fa * b1v;                                     \
    const float ca   = cosf(hal), sa = sinf(hal);                           \
    const float ca2  = ca * ca, sa2 = sa * sa, casa = ca * sa;              \
    const float c2ms2 = ca2 - sa2;                                          \
    const float er = cosf(ph), ei = sinf(ph);                               \
    const float e2r = er * er - ei * ei;                                    \
    const float e2i = 2.0f * er * ei;

// ---- one EPG step: read arrays SP*, write post-shift state into DP* ----
#define EPG_STEP(SP, DP)                                                    \
  {                                                                         \
    float eFpr = 0.f, eFpi = 0.f, eZf = 0.f, eZb = 0.f;                     \
    float eFmr = 0.f, eFmi = 0.f;                                           \
    _Pragma("unroll")                                                       \
    for (int ls = 0; ls < LS; ++ls) {                                       \
      const float ar = SP##Fpr[ls], ai = SP##Fpi[ls];                       \
      const float mr = SP##Fmr[ls], mi = SP##Fmi[ls];                       \
      const float zf = SP##Zf[ls],  zb = SP##Zb[ls];                        \
      /* T2 relax + B0 precession (merged complex scale) */                 \
      const float ar1 = ar * pr - ai * pi_;                                 \
      const float ai1 = ar * pi_ + ai * pr;                                 \
      const float mr1 = mr * pr + mi * pi_;                                 \
      const float mi1 = mi * pr - mr * pi_;                                 \
      /* T1 relax + MT exchange (dZb == -dZf exactly) */                    \
      const float dZ  = kbdt * zb - kfdt * zf;                              \
      const float zf1 = zf * E1f + (ls == 0 ? rffl : 0.0f) + dZ;            \
      const float zb1 = zb * E1b + (ls == 0 ? rfbl : 0.0f) - dZ;            \
      /* RF rotation (free pool only) */                                    \
      const float czf  = casa * zf1;                                        \
      const float nFpr = ca2 * ar1 + sa2 * (mr1 * e2r + mi1 * e2i)          \
                       - czf * ei;                                          \
      const float nFpi = ca2 * ai1 + sa2 * (mr1 * e2i - mi1 * e2r)          \
                       + czf * er;                                          \
      const float nFmr = sa2 * (ar1 * e2r - ai1 * e2i) + ca2 * mr1          \
                       - czf * ei;                                          \
      const float nFmi = -sa2 * (ar1 * e2i + ai1 * e2r) + ca2 * mi1         \
                       - czf * er;                                          \
      const float nZf  = casa * ((ai1 * er - ar1 * ei)                      \
                                 - (mr1 * ei + mi1 * er)) + c2ms2 * zf1;    \
      /* EPG shift folded into destination indices */                       \
      if (ls == LS - 1) { eFpr = nFpr; eFpi = nFpi; eZf = nZf; eZb = zb1; } \
      else { DP##Fpr[ls + 1] = nFpr; DP##Fpi[ls + 1] = nFpi;                \
             DP##Zf[ls + 1]  = nZf;  DP##Zb[ls + 1]  = zb1; }               \
      if (ls == 0) { eFmr = nFmr; eFmi = nFmi; }                            \
      else { DP##Fmr[ls - 1] = nFmr; DP##Fmi[ls - 1] = nFmi; }              \
    }                                                                       \
    /* lane-pair boundary exchange (g=10 <-> g=11) */                       \
    const float bFpr = __shfl_xor(eFpr, 1, 32);                             \
    const float bFpi = __shfl_xor(eFpi, 1, 32);                             \
    const float bZf  = __shfl_xor(eZf,  1, 32);                             \
    const float bZb  = __shfl_xor(eZb,  1, 32);                             \
    const float bFmr = __shfl_xor(eFmr, 1, 32);                             \
    const float bFmi = __shfl_xor(eFmi, 1, 32);                             \
    /* even: Fp/Z order 0 zeroed by shift; odd: order 11 <- partner g10 */  \
    DP##Fpr[0] = odd ? bFpr : 0.f;                                          \
    DP##Fpi[0] = odd ? bFpi : 0.f;                                          \
    DP##Zf[0]  = odd ? bZf  : 0.f;                                          \
    DP##Zb[0]  = odd ? bZb  : 0.f;                                          \
    /* even: Fm order 10 <- partner g11; odd: dummy slot Fm = 0 */          \
    DP##Fmr[LS - 1] = odd ? 0.f : bFmr;                                     \
    DP##Fmi[LS - 1] = odd ? 0.f : bFmi;                                     \
    /* keep odd lane's dummy slot (g=21) identically zero */                \
    DP##Fpr[LS - 1] *= keep10; DP##Fpi[LS - 1] *= keep10;                   \
    DP##Zf[LS - 1]  *= keep10; DP##Zb[LS - 1]  *= keep10;                   \
  }

// ---- stream packed records [Fp.r, Fp.i, Fm.r, Fm.i, Zf, 0,0,0,0, Zb] ----
#define EPG_STORE(SP, pidx)                                                 \
  {                                                                         \
    float* rec = out + (size_t)(pidx) * PB + recbase;                       \
    _Pragma("unroll")                                                       \
    for (int ls = 0; ls < LS - 1; ++ls) {                                   \
      v2f* q = (v2f*)(rec + ls * 10);                                       \
      v2f v0 = { SP##Fpr[ls], SP##Fpi[ls] };                                \
      v2f v1 = { SP##Fmr[ls], SP##Fmi[ls] };                                \
      v2f v2 = { SP##Zf[ls], 0.f };                                         \
      v2f v3 = { 0.f, 0.f };                                                \
      v2f v4 = { 0.f, SP##Zb[ls] };                                         \
      __builtin_nontemporal_store(v0, q + 0);                               \
      __builtin_nontemporal_store(v1, q + 1);                               \
      __builtin_nontemporal_store(v2, q + 2);                               \
      __builtin_nontemporal_store(v3, q + 3);                               \
      __builtin_nontemporal_store(v4, q + 4);                               \
    }                                                                       \
    if (!odd) { /* order g=10 exists only on the even lane */               \
      v2f* q = (v2f*)(rec + (LS - 1) * 10);                                 \
      v2f v0 = { SP##Fpr[LS - 1], SP##Fpi[LS - 1] };                        \
      v2f v1 = { SP##Fmr[LS - 1], SP##Fmi[LS - 1] };                        \
      v2f v2 = { SP##Zf[LS - 1], 0.f };                                     \
      v2f v3 = { 0.f, 0.f };                                                \
      v2f v4 = { 0.f, SP##Zb[LS - 1] };                                     \
      __builtin_nontemporal_store(v0, q + 0);                               \
      __builtin_nontemporal_store(v1, q + 1);                               \
      __builtin_nontemporal_store(v2, q + 2);                               \
      __builtin_nontemporal_store(v3, q + 3);                               \
      __builtin_nontemporal_store(v4, q + 4);                               \
    }                                                                       \
  }

__global__ __launch_bounds__(64) void epg_mt_scan_pair_kernel(
    const float* __restrict__ flip,   // (P,)
    const float* __restrict__ phase,  // (P,)
    const float* __restrict__ T1f, const float* __restrict__ T2f,
    const float* __restrict__ T1b, const float* __restrict__ T2b,
    const float* __restrict__ kf,  const float* __restrict__ kb,
    const float* __restrict__ B0,  const float* __restrict__ B1,
    const float* __restrict__ wf,  const float* __restrict__ wb,
    const int*   __restrict__ TRp,
    float* __restrict__ out,          // (P, B, 21, 10)
    int n_pulses, int B)
{
  const int tid = blockIdx.x * blockDim.x + threadIdx.x;
  const int vb  = tid >> 1;          // voxel index
  const int odd = tid & 1;           // 0: orders 0..10, 1: orders 11..20
  if (vb >= B) return;

  const float TR = (float)TRp[0];
  const float dt = TR * 0.001f;

  // ---- per-voxel constants (both lanes of a pair load the same voxel) ----
  const float E1f  = expf(-TR / T1f[vb]);
  const float E2f  = expf(-TR / T2f[vb]);
  const float E1b  = expf(-TR / T1b[vb]);
  (void)T2b;  // only ever scales the identically-zero bound F states
  const float kfdt = kf[vb] * dt;
  const float kbdt = kb[vb] * dt;
  const float wfv  = wf[vb];
  const float wbv  = wb[vb];
  const float b1v  = B1[vb];
  // T1 recovery feed exists only at global order 0 (even lane, slot 0)
  const float rffl = odd ? 0.f : (1.0f - E1f) * wfv;
  const float rfbl = odd ? 0.f : (1.0f - E1b) * wbv;
  const float keep10 = odd ? 0.f : 1.f;   // dummy-slot mask

  const float phi = 6.2831853071795864769f * B0[vb] * TR * 0.001f;
  const float b0c = cosf(phi);
  const float b0s = sinf(phi);
  const float pr  = E2f * b0c;   // Fpf *= (pr + i*pi_), Fmf *= (pr - i*pi_)
  const float pi_ = E2f * b0s;

  // ---- ping-pong state in registers ----
  float AFpr[LS], AFpi[LS], AFmr[LS], AFmi[LS], AZf[LS], AZb[LS];
  float BFpr[LS], BFpi[LS], BFmr[LS], BFmi[LS], BZf[LS], BZb[LS];
#pragma unroll
  for (int ls = 0; ls < LS; ++ls) {
    AFpr[ls] = 0.f; AFpi[ls] = 0.f; AFmr[ls] = 0.f; AFmi[ls] = 0.f;
    AZf[ls] = 0.f;  AZb[ls] = 0.f;
    BFpr[ls] = 0.f; BFpi[ls] = 0.f; BFmr[ls] = 0.f; BFmi[ls] = 0.f;
    BZf[ls] = 0.f;  BZb[ls] = 0.f;
  }
  AZf[0] = odd ? 0.f : wfv;   // Zf[0] = wf lives at global order 0
  AZb[0] = odd ? 0.f : wbv;

  const size_t PB      = (size_t)B * (N_STATES * 10);
  const size_t recbase = (size_t)vb * (N_STATES * 10) + (odd ? LS * 10 : 0);

  for (int p = 0; p < n_pulses; p += 2) {
    {
      EPG_PULSE_CONSTS(p);
      EPG_STEP(A, B);
      EPG_STORE(B, p);
    }
    if (p + 1 < n_pulses) {
      EPG_PULSE_CONSTS(p + 1);
      EPG_STEP(B, A);
      EPG_STORE(A, p + 1);
    }
  }
}

extern "C" void kernel_launch(void* const* d_in, const int* in_sizes, int n_in,
                              void* d_out, int out_size, void* d_ws, size_t ws_size,
                              hipStream_t stream) {
  // input order: flip_angles, phases, T1f, T2f, T1b, T2b, kf, kb, B0, B1,
  //              wf, wb, TR, TE
  const float* flip  = (const float*)d_in[0];
  const float* phase = (const float*)d_in[1];
  const float* T1f   = (const float*)d_in[2];
  const float* T2f   = (const float*)d_in[3];
  const float* T1b   = (const float*)d_in[4];
  const float* T2b   = (const float*)d_in[5];
  const float* kf    = (const float*)d_in[6];
  const float* kb    = (const float*)d_in[7];
  const float* B0    = (const float*)d_in[8];
  const float* B1    = (const float*)d_in[9];
  const float* wf    = (const float*)d_in[10];
  const float* wb    = (const float*)d_in[11];
  const int*   TRp   = (const int*)d_in[12];
  // d_in[13] = TE, unused by the reference simulation.

  const int n_pulses = in_sizes[0];
  const int B        = in_sizes[2];

  const int threads = 2 * B;                     // one lane pair per voxel
  const int block   = 64;                        // 2 waves per workgroup
  const int grid    = (threads + block - 1) / block;
  epg_mt_scan_pair_kernel<<<grid, block, 0, stream>>>(
      flip, phase, T1f, T2f, T1b, T2b, kf, kb, B0, B1, wf, wb, TRp,
      (float*)d_out, n_pulses, B);
}